// PatchTransformer_61503931679130
// MI455X (gfx1250) — compile-verified
//
#include <hip/hip_runtime.h>
#include <stdint.h>

// MI455X / gfx1250. This op is a translated bilinear patch-paste + composite:
// ~30 MB of unique traffic -> ~1.3 us at 23.3 TB/s HBM (L2-resident anyway).
// No matmul structure => WMMA inapplicable; we exercise the CDNA5 data-movement
// paths instead: TDM tensor_load_to_lds (+ s_wait_tensorcnt) to stage the
// broadcast patch_locations into LDS, global_prefetch_b8 for patch rows, and
// b128 float4 vectorization for the dominant image read/write traffic.

typedef __attribute__((ext_vector_type(4))) unsigned int u32x4;
typedef __attribute__((ext_vector_type(4))) int          i32x4;
typedef __attribute__((ext_vector_type(8))) int          i32x8;

#define IMS    1024
#define NPATCH 8
#define PDIM   256
#define NCH    3

__global__ __launch_bounds__(256) void patch_composite_kernel(
    const float* __restrict__ patches,   // (8,3,256,256)
    const float* __restrict__ locs,      // (8,2)
    const float* __restrict__ img,       // (3,1024,1024)
    float*       __restrict__ out)       // (3,1024,1024)
{
#pragma clang fp contract(off)
  __shared__ float s_loc[2 * NPATCH];

  // ---- TDM: DMA the 16 location floats into LDS (1-D tensor descriptor) ----
  if (threadIdx.x == 0) {
    uint64_t ga  = (uint64_t)(uintptr_t)locs;
    uint32_t lds = (uint32_t)(uintptr_t)&s_loc[0];  // low 32 bits of generic ptr = LDS byte addr
    u32x4 g0;
    g0[0] = 1u;                                             // count=1, user D#
    g0[1] = lds;                                            // lds_addr
    g0[2] = (uint32_t)ga;                                   // global_addr[31:0]
    g0[3] = ((uint32_t)(ga >> 32) & 0x01FFFFFFu) | (2u << 30); // addr[56:32] | type=2
    i32x8 g1;
    g1[0] = 0x00020000;        // data_size=2 (4 bytes), no multicast/pad/iterate
    g1[1] = (int)(16u << 16);  // tensor_dim0 = 16
    g1[2] = (int)(1u  << 16);  // tensor_dim1 = 1
    g1[3] = (int)(16u << 16);  // tile_dim0 = 16
    g1[4] = 0;                 // tile_dim1 = 0 (unused), tile_dim2 = 0
    g1[5] = 16;                // tensor_dim0_stride = 16
    g1[6] = 0;
    g1[7] = 0;
    i32x4 z4 = (i32x4){0, 0, 0, 0};
#if defined(__clang_major__) && (__clang_major__ >= 23)
    i32x8 z8 = (i32x8){0, 0, 0, 0, 0, 0, 0, 0};
    __builtin_amdgcn_tensor_load_to_lds(g0, g1, z4, z4, z8, 0);   // 6-arg (clang-23)
#else
    __builtin_amdgcn_tensor_load_to_lds(g0, g1, z4, z4, 0);       // 5-arg (ROCm 7.2 / clang-22)
#endif
    __builtin_amdgcn_s_wait_tensorcnt(0);
  }
  __syncthreads();

  // ---- 4 pixels per thread along w (float4 = b128 traffic) ----
  const int tid = blockIdx.x * 256 + threadIdx.x;   // 0 .. 262143
  const int w0  = (tid << 2) & (IMS - 1);
  const int h   = (tid << 2) >> 10;

  const float4* img4  = reinterpret_cast<const float4*>(img);
  float4*       out4  = reinterpret_cast<float4*>(out);
  const int plane4    = (IMS * IMS) >> 2;
  const int idx4      = (h * IMS + w0) >> 2;

  float pix[NCH][4];
  for (int c = 0; c < NCH; ++c) {
    float4 v = img4[idx4 + c * plane4];
    pix[c][0] = v.x; pix[c][1] = v.y; pix[c][2] = v.z; pix[c][3] = v.w;
  }

  // gy exact: (2h+1)/1024 - 1 (all steps exact in f32, matches reference)
  const float gy = (float)(2 * h + 1) * (1.0f / 1024.0f) - 1.0f;

  for (int n = 0; n < NPATCH; ++n) {
    const float lx = s_loc[2 * n + 0];
    const float ly = s_loc[2 * n + 1];

    const float gry = gy - ly;
    const float yy  = ((gry + 1.0f) * 1024.0f - 1.0f) * 0.5f;
    const float y0f = floorf(yy);
    if (y0f < -1.0f || y0f > 255.0f) continue;   // all taps sample zero -> background
    const float wy1 = yy - y0f;
    const float wy0 = 1.0f - wy1;
    const int   y0  = (int)y0f;
    const int   y1  = y0 + 1;
    const float vy0 = (y0 >= 0) ? 1.0f : 0.0f;   // image-bounds validity of row y0
    const bool  by0 = (y0 >= 0);                  // patch-extent (y0 < 256 given bbox)
    const bool  by1 = (y1 < PDIM);

    const float* pb = patches + (size_t)n * NCH * PDIM * PDIM;
    __builtin_prefetch(pb + (size_t)(by0 ? y0 : 0) * PDIM, 0, 1);  // global_prefetch_b8

    const size_t r0 = (size_t)(by0 ? y0 : 0) * PDIM;
    const size_t r1 = (size_t)(by1 ? y1 : 0) * PDIM;

    for (int j = 0; j < 4; ++j) {
      const int   w   = w0 + j;
      const float gx  = (float)(2 * w + 1) * (1.0f / 1024.0f) - 1.0f;
      const float grx = gx - lx;
      const float xx  = ((grx + 1.0f) * 1024.0f - 1.0f) * 0.5f;
      const float x0f = floorf(xx);
      if (x0f < -1.0f || x0f > 255.0f) continue;
      const float wx1 = xx - x0f;
      const float wx0 = 1.0f - wx1;
      const int   x0  = (int)x0f;
      const int   x1  = x0 + 1;
      const float vx0 = (x0 >= 0) ? 1.0f : 0.0f;
      const bool  bx0 = (x0 >= 0);
      const bool  bx1 = (x1 < PDIM);

      const float w00 = wy0 * wx0;
      const float w01 = wy0 * wx1;
      const float w10 = wy1 * wx0;
      const float w11 = wy1 * wx1;

      // mask_w = bilinear of ones with image-bounds validity; exact ref ordering
      float m = (vy0 * vx0) * w00;
      m = m + vy0 * w01;     // (y0, x1): x1 always in-image here
      m = m + vx0 * w10;     // (y1, x0): y1 always in-image here
      m = m + w11;           // (y1, x1): always valid
      if (m != 1.0f) continue;   // reference zeroes non-exact mask -> pv==0 -> keep bg

      const float f00 = (by0 && bx0) ? 1.0f : 0.0f;   // patch-extent validity (pad=0)
      const float f01 = (by0 && bx1) ? 1.0f : 0.0f;
      const float f10 = (by1 && bx0) ? 1.0f : 0.0f;
      const float f11 = (by1 && bx1) ? 1.0f : 0.0f;
      const int   cx0 = bx0 ? x0 : 0;                 // clipped (matches jnp.clip gather)
      const int   cx1 = bx1 ? x1 : 0;

      for (int c = 0; c < NCH; ++c) {
        const float* pc = pb + c * (PDIM * PDIM);
        float s = (pc[r0 + cx0] * f00) * w00;
        s = s + (pc[r0 + cx1] * f01) * w01;
        s = s + (pc[r1 + cx0] * f10) * w10;
        s = s + (pc[r1 + cx1] * f11) * w11;
        const float pv = s * m;                       // m == 1.0 here
        pix[c][j] = (pv == 0.0f) ? pix[c][j] : pv;    // where(adv==0, img, adv)
      }
    }
  }

  for (int c = 0; c < NCH; ++c) {
    float4 v;
    v.x = pix[c][0]; v.y = pix[c][1]; v.z = pix[c][2]; v.w = pix[c][3];
    out4[idx4 + c * plane4] = v;
  }
}

extern "C" void kernel_launch(void* const* d_in, const int* in_sizes, int n_in,
                              void* d_out, int out_size, void* d_ws, size_t ws_size,
                              hipStream_t stream) {
  (void)in_sizes; (void)n_in; (void)out_size; (void)d_ws; (void)ws_size;
  const float* patches = (const float*)d_in[0];   // adv_patches (8,3,256,256) f32
  const float* locs    = (const float*)d_in[1];   // patch_locations (8,2) f32
  const float* img     = (const float*)d_in[2];   // img_clean (3,1024,1024) f32
  float* out = (float*)d_out;                     // (1,3,1024,1024) f32

  const int total_threads = (IMS * IMS) / 4;      // 4 pixels per thread
  patch_composite_kernel<<<dim3(total_threads / 256), dim3(256), 0, stream>>>(
      patches, locs, img, out);
}